// Denoiser_10136122818921
// MI455X (gfx1250) — compile-verified
//
#include <hip/hip_runtime.h>

typedef __attribute__((ext_vector_type(16))) _Float16 v16h;
typedef __attribute__((ext_vector_type(8)))  _Float16 v8h;

union VH  { int i[8]; v16h v; };
union VH4 { int i[4]; v8h  v; };

__device__ __forceinline__ int pkrtz(float a, float b) {
    return __builtin_bit_cast(int, __builtin_amdgcn_cvt_pkrtz(a, b)); // v_cvt_pk_rtz_f16_f32
}

#define T 8  // tiles (16 points each) per wave: 8 independent WMMA chains

// D(f16 16x16) -> B(f16 32x16, K-padded) repack with fused ReLU.
// D VGPR v = packed rows (2v,2v+1) @ lanes<16, rows (2v+8,2v+9) @ lanes>=16.
// B[v] wants the former in lanes<16 and the latter in lanes>=16 (table-order A);
// B[v+4] may hold any finite value (it hits A's zero elements 8..15).
__device__ __forceinline__ void repack_relu(VH& Bt, const VH4& Dt) {
#pragma unroll
    for (int v = 0; v < 4; ++v) {
        int lo;
        asm("v_pk_max_num_f16 %0, %1, 0" : "=v"(lo) : "v"(Dt.i[v])); // ReLU, no canonicalize
        int hi = lo;
        asm("v_permlane16_swap_b32 %0, %1" : "+v"(lo), "+v"(hi));    // pure-VALU half swap
        Bt.i[v]     = lo;   // lanes<16 = rows 2v,2v+1
        Bt.i[v + 4] = hi;   // lanes<16 = rows 2v+8,2v+9
    }
}

__global__ __launch_bounds__(256) void mlp16_wmma(
    const float* __restrict__ x, const float* __restrict__ w_in,
    const float* __restrict__ w_mid, const float* __restrict__ w_out,
    float* __restrict__ out, int n)
{
    const int lane = threadIdx.x & 31;
    const int wave = (blockIdx.x * blockDim.x + threadIdx.x) >> 5;
    const int row  = lane & 15;              // A-matrix row / point-in-tile
    const int ks   = (lane < 16) ? 0 : 8;    // K sub-range for f16 A layout

    // ---- 6 hidden-layer A-operands: [0]=w_in (K=2, padded), [1..5]=w_mid ----
    VH WA[6];
    WA[0].i[0] = (lane < 16) ? pkrtz(w_in[2*row], w_in[2*row + 1]) : 0;  // K=0,1
#pragma unroll
    for (int j = 1; j < 8; ++j) WA[0].i[j] = 0;                          // K=2..31 zero
#pragma unroll
    for (int m = 0; m < 5; ++m) {
        const float* Wp = w_mid + m * 256 + row * 16 + ks;
#pragma unroll
        for (int j = 0; j < 4; ++j) WA[m + 1].i[j] = pkrtz(Wp[2*j], Wp[2*j+1]);
#pragma unroll
        for (int j = 4; j < 8; ++j) WA[m + 1].i[j] = 0;
    }
    // ---- w_out padded to 16x32 A-operand (rows 2..15 zero) ----
    VH Wo;
#pragma unroll
    for (int j = 0; j < 8; ++j) Wo.i[j] = 0;
    if (row < 2) {
        const float* Wp = w_out + row * 16 + ks;
#pragma unroll
        for (int j = 0; j < 4; ++j) Wo.i[j] = pkrtz(Wp[2*j], Wp[2*j+1]);
    }

    const int base = wave * (16 * T);

    // ---- B for layer 0: V0 = pack(x0,x1) per point; V1..7 = finite junk ----
    VH B[T];
#pragma unroll
    for (int t = 0; t < T; ++t) {
        int p  = base + t * 16 + row;
        int pc = (p < n) ? p : 0;
        float2 xv = ((const float2*)x)[pc];
        int pk0 = pkrtz(xv.x, xv.y);             // K=0,1 @ point
#pragma unroll
        for (int j = 0; j < 8; ++j) B[t].i[j] = pk0;   // j>0: hits zero A-rows
    }

    // ---- 6 hidden layers: f16 WMMA + fused ReLU/repack (D layout IS B layout) ----
#pragma unroll
    for (int m = 0; m < 6; ++m) {
        VH4 D[T];
#pragma unroll
        for (int t = 0; t < T; ++t) {
            v8h c = {};
            D[t].v = __builtin_amdgcn_wmma_f16_16x16x32_f16(
                         false, WA[m].v, false, B[t].v, (short)0, c, false, false);
        }
#pragma unroll
        for (int t = 0; t < T; ++t) repack_relu(B[t], D[t]);
    }

    // ---- output layer: D VGPR0 lanes 0..15 = packed (out0,out1) per point ----
    VH4 Do[T];
#pragma unroll
    for (int t = 0; t < T; ++t) {
        v8h c = {};
        Do[t].v = __builtin_amdgcn_wmma_f16_16x16x32_f16(
                      false, Wo.v, false, B[t].v, (short)0, c, false, false);
    }
    if (lane < 16) {
#pragma unroll
        for (int t = 0; t < T; ++t) {
            int p = base + t * 16 + lane;
            if (p < n) {
                float2 o;
                o.x = (float)Do[t].v[0];   // M=0: out feature 0
                o.y = (float)Do[t].v[1];   // M=1: out feature 1
                ((float2*)out)[p] = o;
            }
        }
    }
}

extern "C" void kernel_launch(void* const* d_in, const int* in_sizes, int n_in,
                              void* d_out, int out_size, void* d_ws, size_t ws_size,
                              hipStream_t stream) {
    const float* x     = (const float*)d_in[0];
    const float* w_in  = (const float*)d_in[1];
    const float* w_mid = (const float*)d_in[2];
    const float* w_out = (const float*)d_in[3];
    float* out = (float*)d_out;
    int n = in_sizes[0] / 2;                     // number of points
    int ptsPerBlock = 256 / 32 * 16 * T;         // 8 waves * 128 pts = 1024
    int blocks = (n + ptsPerBlock - 1) / ptsPerBlock;
    mlp16_wmma<<<blocks, 256, 0, stream>>>(x, w_in, w_mid, w_out, out, n);
}